// DelayedXOR_SH_SNN_2637109920461
// MI455X (gfx1250) — compile-verified
//
#include <hip/hip_runtime.h>

typedef __attribute__((ext_vector_type(2))) float v2f;
typedef __attribute__((ext_vector_type(8))) float v8f;

#define B_DIM 1024
#define T_DIM 2048
#define I_DIM 16
#define H_DIM 64
#define TT    16              // time tile (WMMA M)
#define SX_STRIDE  17         // padded x-tile row stride (bank-conflict-free A reads)
#define SIC_STRIDE 68         // padded ic-tile row stride (conflict-free D writes + scan reads)

__global__ __launch_bounds__(128)
void snn_fused_kernel(const float* __restrict__ x,
                      const float* __restrict__ W1,
                      const float* __restrict__ b1,
                      const float* __restrict__ tau_m,
                      const float* __restrict__ W2,
                      const float* __restrict__ b2,
                      float* __restrict__ out)
{
    __shared__ float sx[TT * SX_STRIDE];    // 16x16 x tile (padded)
    __shared__ float sic[TT * SIC_STRIDE];  // 16t x 64h ic tile (padded)
    __shared__ float sred[H_DIM];

    const int tid  = threadIdx.x;
    const int wave = tid >> 5;       // 4 waves, wave w -> h slice [16w, 16w+16)
    const int lane = tid & 31;
    const int half = lane >> 4;      // ISA: lanes 16-31 carry K+2
    const int l16  = lane & 15;
    const int b    = blockIdx.x;

    // ---- B fragment: W1^T (K=16 x N=16 per wave), loop invariant ----
    // 16x16x4 f32 layout: VGPR v, lane L -> K = 2*(L/16) + v ; N = L%16
    const int hB = wave * 16 + l16;
    v2f bfrag[4];
#pragma unroll
    for (int j = 0; j < 4; ++j) {
        const int k = 4 * j + 2 * half;
        bfrag[j].x = W1[hB * I_DIM + k];
        bfrag[j].y = W1[hB * I_DIM + k + 1];
    }
    const float biasB = b1[hB];  // same bias down every column N -> fold into C init

    // ---- recurrence state: threads 0..63 own h = tid ----
    float alpha = 0.f, vmem = 0.f, spk = 0.f, cnt = 0.f, w2h = 0.f;
    if (tid < H_DIM) {
        alpha = 1.0f / (1.0f + __expf(-tau_m[tid]));  // sigmoid
        w2h   = W2[tid];                              // W2 is [1, H]
    }
    const float oma = 1.0f - alpha;

    const float* xb = x + (size_t)b * T_DIM * I_DIM;

    // staging: 128 threads x float2 = 256 floats = one 16x16 tile
    const int srow = tid >> 3;
    const int scol = (tid & 7) * 2;

    for (int t0 = 0; t0 < T_DIM; t0 += TT) {
        // ---- stage x tile: coalesced global_load_b64, scalar ds stores (padded) ----
        const float* gp = xb + (size_t)(t0 + srow) * I_DIM + scol;
        v2f xv = *(const v2f*)gp;
        sx[srow * SX_STRIDE + scol]     = xv.x;
        sx[srow * SX_STRIDE + scol + 1] = xv.y;
        if (t0 + TT < T_DIM) {
            __builtin_prefetch(gp + TT * I_DIM, 0, 1);  // global_prefetch_b8 next tile
        }
        __syncthreads();

        // ---- ic tile = x_tile(16x16) @ W1^T slice(16x16) + b1, fp32 WMMA, K=16 ----
        v8f acc;
#pragma unroll
        for (int r = 0; r < 8; ++r) acc[r] = biasB;
#pragma unroll
        for (int j = 0; j < 4; ++j) {
            const int k = 4 * j + 2 * half;
            v2f a;
            a.x = sx[l16 * SX_STRIDE + k];
            a.y = sx[l16 * SX_STRIDE + k + 1];
            acc = __builtin_amdgcn_wmma_f32_16x16x4_f32(
                false, a, false, bfrag[j], (short)0, acc, false, false);
        }
        // D layout: VGPR r, lane L -> M = r + 8*(L/16), N = L%16
#pragma unroll
        for (int r = 0; r < 8; ++r) {
            sic[(r + 8 * half) * SIC_STRIDE + wave * 16 + l16] = acc[r];
        }
        __syncthreads();

        // ---- LIF recurrence over the 16 timesteps (V_TH = 1.0) ----
        if (tid < H_DIM) {
            if (t0 >= T_DIM / 2) {
#pragma unroll
                for (int tl = 0; tl < TT; ++tl) {
                    const float ic = sic[tl * SIC_STRIDE + tid];
                    vmem = alpha * vmem + oma * ic - spk;
                    spk  = (vmem > 1.0f) ? 1.0f : 0.0f;
                    cnt += spk;
                }
            } else {
#pragma unroll
                for (int tl = 0; tl < TT; ++tl) {
                    const float ic = sic[tl * SIC_STRIDE + tid];
                    vmem = alpha * vmem + oma * ic - spk;
                    spk  = (vmem > 1.0f) ? 1.0f : 0.0f;
                }
            }
        }
        // no barrier needed here: next iteration only touches sx before the
        // post-stage barrier, and sic is rewritten only after that barrier.
    }

    // ---- out[b] = sum_h cnt[h] * W2[h] + b2 ----
    if (tid < H_DIM) sred[tid] = cnt * w2h;
    __syncthreads();
    if (tid == 0) {
        float a0 = b2[0];
#pragma unroll
        for (int i = 0; i < H_DIM; ++i) a0 += sred[i];
        out[b] = a0;
    }
}

extern "C" void kernel_launch(void* const* d_in, const int* in_sizes, int n_in,
                              void* d_out, int out_size, void* d_ws, size_t ws_size,
                              hipStream_t stream) {
    const float* x     = (const float*)d_in[0];
    const float* W1    = (const float*)d_in[1];
    const float* b1    = (const float*)d_in[2];
    const float* tau_m = (const float*)d_in[3];
    const float* W2    = (const float*)d_in[4];
    const float* b2    = (const float*)d_in[5];
    float* out = (float*)d_out;

    const int Bn = in_sizes[0] / (T_DIM * I_DIM);  // 1024
    snn_fused_kernel<<<Bn, 128, 0, stream>>>(x, W1, b1, tau_m, W2, b2, out);
}